// DiscriminativeLoss_22471268893125
// MI455X (gfx1250) — compile-verified
//
#include <hip/hip_runtime.h>
#include <hip/hip_bf16.h>

typedef float v2f __attribute__((ext_vector_type(2)));
typedef float v4f __attribute__((ext_vector_type(4)));
typedef float v8f __attribute__((ext_vector_type(8)));
typedef unsigned int u32x4 __attribute__((ext_vector_type(4)));
typedef int i32x4 __attribute__((ext_vector_type(4)));
typedef int i32x8 __attribute__((ext_vector_type(8)));

// Problem dims (fixed by reference)
#define BB 4
#define DD 32
#define HH 512
#define WW 1024
#define HWP (HH * WW)   // 524288
#define CC 8

#define TILE 256
#define PITCH (TILE + 2)   // TDM pad: +2 DWORDs per 256 DWORDs -> conflict-free b64 LDS reads
#define NBLK1 256          // blocks per batch, pass 1
#define NBLK2 512          // blocks per batch, pass 2

// workspace layout (float offsets)
#define WS_MUACC 0        // [B][16][32] padded-cluster mu numerators = 2048
#define WS_MSUM  2048     // [B][8]  mask counts                       = 32
#define WS_MU    2080     // [B][8][32] final mu                       = 1024
#define WS_MUSQ  3104     // [B][8]  ||mu||^2                          = 32
#define WS_DIST  3136     // [B]                                        = 4
#define WS_REG   3140     // [B]                                        = 4
#define WS_VAR   3144     // [B][8] hinge^2 pixel sums                 = 32
#define WS_ZERO_N 4096

__device__ __forceinline__ void atomAddG(float* p, float v) {
    __hip_atomic_fetch_add(p, v, __ATOMIC_RELAXED, __HIP_MEMORY_SCOPE_AGENT);
}
__device__ __forceinline__ void atomAddL(float* p, float v) {
    __hip_atomic_fetch_add(p, v, __ATOMIC_RELAXED, __HIP_MEMORY_SCOPE_WORKGROUP);
}

// ---------------------------------------------------------------------------
// Tensor Data Mover: 2D tile load (rows x TILE elements, 4-byte elements,
// row stride HWP elements) from global to LDS with pad_interval=256 DW,
// pad_amount=2 DW so the LDS pitch becomes PITCH = 258 floats.
// D# encoding per CDNA5 ISA ch.8 (groups 2/3 zero => 2-D tensor).
// ---------------------------------------------------------------------------
__device__ __forceinline__ void tdm_load_2d(unsigned ldsOff, const void* gptr, unsigned rows) {
    unsigned long long ga = (unsigned long long)(uintptr_t)gptr;
    u32x4 g0;
    g0[0] = 1u;                                             // count=1, user descriptor
    g0[1] = ldsOff;                                         // lds_addr (bytes)
    g0[2] = (unsigned)ga;                                   // global_addr[31:0]
    g0[3] = (unsigned)((ga >> 32) & 0x1FFFFFFu) | (2u << 30); // global_addr[56:32] | type=2
    i32x8 g1;
    g1[0] = (int)((2u << 16)        // data_size = 4 bytes
                | (1u << 20)        // pad_enable
                | (7u << 22)        // pad_interval: 256 DWORDs
                | (1u << 25));      // pad_amount: 2 DWORDs
    g1[1] = (int)((HWP & 0xFFFFu) << 16);                   // tensor_dim0[15:0]
    g1[2] = (int)(((unsigned)HWP >> 16) | (rows << 16));    // tensor_dim0[31:16] | tensor_dim1[15:0]
    g1[3] = (int)((unsigned)TILE << 16);                    // tile_dim0
    g1[4] = (int)rows;                                      // tile_dim1 (tile_dim2 = 0)
    g1[5] = (int)HWP;                                       // tensor_dim0_stride[31:0]
    g1[6] = 0;
    g1[7] = 0;
    i32x4 z4 = {0, 0, 0, 0};
#if __clang_major__ >= 23
    i32x8 z8 = {0, 0, 0, 0, 0, 0, 0, 0};
    __builtin_amdgcn_tensor_load_to_lds(g0, g1, z4, z4, z8, 0);
#else
    __builtin_amdgcn_tensor_load_to_lds(g0, g1, z4, z4, 0);
#endif
}

__global__ __launch_bounds__(256) void dl_zero_kernel(float* ws) {
    int i = blockIdx.x * 256 + threadIdx.x;
    if (i < WS_ZERO_N) ws[i] = 0.0f;
}

// ---------------------------------------------------------------------------
// Pass 1: mu numerators via V_WMMA_F32_16X16X4_F32 over TDM-staged,
// double-buffered LDS tiles; mask counts from staged ints.
// ---------------------------------------------------------------------------
__global__ __launch_bounds__(256) void dl_pass1_kernel(const float* __restrict__ E,
                                                       const int* __restrict__ M,
                                                       float* __restrict__ muAcc,
                                                       float* __restrict__ msumAcc) {
    __shared__ __align__(16) float eBuf[2][DD * PITCH];  // 2 x 33024 B
    __shared__ __align__(16) int   mBuf[2][CC * PITCH];  // 2 x 8256 B
    __shared__ float muRed[16 * 32];
    __shared__ float msumSh[CC];

    const int tx   = threadIdx.x;
    const int b    = blockIdx.x >> 8;      // NBLK1 == 256
    const int blk  = blockIdx.x & (NBLK1 - 1);
    const int wave = tx >> 5;
    const int lane = tx & 31;
    const int half = lane >> 4;            // K lane-half
    const int lr   = lane & 15;            // A: cluster row  /  B: d column

    const float* Eb = E + (long long)b * DD * HWP;
    const int*   Mb = M + (long long)b * CC * HWP;

    float msumReg[CC];
#pragma unroll
    for (int c = 0; c < CC; ++c) msumReg[c] = 0.0f;
    v8f acc0 = {};  // D tile: d = 0..15
    v8f acc1 = {};  // D tile: d = 16..31

    const int NT = (HWP / TILE) / NBLK1;   // 8 tiles per block

    if (wave == 0) {  // prefetch tile 0
        const int p0 = blk * TILE;
        tdm_load_2d((unsigned)(uintptr_t)&eBuf[0][0], Eb + p0, DD);
        tdm_load_2d((unsigned)(uintptr_t)&mBuf[0][0], Mb + p0, CC);
    }

    for (int it = 0; it < NT; ++it) {
        const int buf = it & 1;
        if (wave == 0) {
            if (it + 1 < NT) {  // prefetch tile it+1 into the other buffer
                const int pn = (blk + (it + 1) * NBLK1) * TILE;
                tdm_load_2d((unsigned)(uintptr_t)&eBuf[buf ^ 1][0], Eb + pn, DD);
                tdm_load_2d((unsigned)(uintptr_t)&mBuf[buf ^ 1][0], Mb + pn, CC);
                __builtin_amdgcn_s_wait_tensorcnt(2);  // tile `it` landed; prefetch in flight
            } else {
                __builtin_amdgcn_s_wait_tensorcnt(0);
            }
        }
        __syncthreads();  // tile `it` visible to all waves

        // mask counts from the staged i32 tile
#pragma unroll
        for (int c = 0; c < CC; ++c) msumReg[c] += (float)mBuf[buf][c * PITCH + tx];

        // wave w handles pixels [32w, 32w+32): 8 K-steps of 4 pixels.
        // A(16x4 f32): lane%16 = M(cluster), K = 2*half + vgpr -> pixels p+2h+{0,1}
        // B(4x16 f32): lane%16 = N(d), same K coords.
        const int pw = wave * 32 + 2 * half;
#pragma unroll
        for (int s = 0; s < 8; ++s) {
            const int p = pw + s * 4;
            v2f a = {0.0f, 0.0f};
            if (lr < CC) {
                const int* mp = &mBuf[buf][lr * PITCH + p];
                a.x = (float)mp[0];
                a.y = (float)mp[1];
            }
            v2f b0 = *(const v2f*)&eBuf[buf][lr * PITCH + p];
            v2f b1 = *(const v2f*)&eBuf[buf][(16 + lr) * PITCH + p];
            acc0 = __builtin_amdgcn_wmma_f32_16x16x4_f32(false, a, false, b0,
                                                         (short)0, acc0, false, false);
            acc1 = __builtin_amdgcn_wmma_f32_16x16x4_f32(false, a, false, b1,
                                                         (short)0, acc1, false, false);
        }
        __syncthreads();  // all waves done reading `buf` before TDM reuses it (tile it+2)
    }

    // Cross-wave reduction of accumulator fragments in LDS, then global atomics.
    if (tx < CC) msumSh[tx] = 0.0f;
    for (int i = tx; i < 16 * 32; i += 256) muRed[i] = 0.0f;
    __syncthreads();
    // C/D fragment: vgpr r, lane L -> M(cluster) = r + 8*half, N(d) = lr
#pragma unroll
    for (int r = 0; r < 8; ++r) {
        atomAddL(&muRed[(r + 8 * half) * 32 + lr], acc0[r]);
        atomAddL(&muRed[(r + 8 * half) * 32 + 16 + lr], acc1[r]);
    }
#pragma unroll
    for (int c = 0; c < CC; ++c) atomAddL(&msumSh[c], msumReg[c]);
    __syncthreads();
    for (int i = tx; i < 16 * 32; i += 256) atomAddG(&muAcc[b * 512 + i], muRed[i]);
    if (tx < CC) atomAddG(&msumAcc[b * CC + tx], msumSh[tx]);
}

// Finalize mu, ||mu||^2, pairwise distance loss, reg loss. One block.
__global__ __launch_bounds__(256) void dl_mu_kernel(float* __restrict__ ws) {
    __shared__ __align__(16) float muL[BB * CC * DD];  // 1024
    __shared__ float distSh[BB];
    __shared__ float regSh[BB];
    const int t = threadIdx.x;
    if (t < BB) { distSh[t] = 0.0f; regSh[t] = 0.0f; }
    __syncthreads();
    if (t < BB * CC) {
        const int b = t >> 3, c = t & 7;
        const float inv = 1.0f / ws[WS_MSUM + t];
        float sq = 0.0f;
#pragma unroll
        for (int d = 0; d < DD; ++d) {
            float m = ws[WS_MUACC + (b * 16 + c) * 32 + d] * inv;
            muL[t * DD + d] = m;
            ws[WS_MU + t * DD + d] = m;
            sq = fmaf(m, m, sq);
        }
        ws[WS_MUSQ + t] = sq;
        atomAddL(&regSh[b], sqrtf(sq + 1e-12f) * (1.0f / CC));
    }
    __syncthreads();
    {   // t encodes (b, i, j): 4*8*8 = 256
        const int b = t >> 6, i = (t >> 3) & 7, j = t & 7;
        if (i < j) {
            float s = 0.0f;
#pragma unroll
            for (int d = 0; d < DD; ++d) {
                float df = muL[(b * CC + i) * DD + d] - muL[(b * CC + j) * DD + d];
                s = fmaf(df, df, s);
            }
            float r = fmaxf(1.5f - sqrtf(s + 1e-12f), 0.0f);
            atomAddL(&distSh[b], r * r);
        }
    }
    __syncthreads();
    if (t < BB) {
        ws[WS_DIST + t] = distSh[t] / (CC * (CC - 1) / 2.0f);
        ws[WS_REG + t] = regSh[t];
    }
}

// Pass 2: per-pixel hinge^2 variance term.
__global__ __launch_bounds__(256) void dl_pass2_kernel(const float* __restrict__ E,
                                                       const int* __restrict__ M,
                                                       const float* __restrict__ mu,
                                                       const float* __restrict__ muSq,
                                                       float* __restrict__ varAcc) {
    __shared__ __align__(16) float muSh[CC * DD];
    __shared__ float muSqSh[CC];
    __shared__ float varSh[CC];
    const int tx = threadIdx.x;
    const int b = blockIdx.x >> 9;  // NBLK2 == 512
    const int blk = blockIdx.x & (NBLK2 - 1);
    muSh[tx] = mu[b * CC * DD + tx];
    if (tx < CC) { muSqSh[tx] = muSq[b * CC + tx]; varSh[tx] = 0.0f; }
    __syncthreads();

    const float* Eb = E + (long long)b * DD * HWP;
    const int* Mb = M + (long long)b * CC * HWP;
    const int g = blk * 256 + tx;  // 131072 threads per batch
    float vr[CC];
#pragma unroll
    for (int c = 0; c < CC; ++c) vr[c] = 0.0f;

    for (int i = 0; i < HWP / (NBLK2 * 256); ++i) {
        const int p = g + i * (NBLK2 * 256);
        float esq = 0.0f;
        float cr[CC];
#pragma unroll
        for (int c = 0; c < CC; ++c) cr[c] = 0.0f;
#pragma unroll
        for (int d4 = 0; d4 < DD; d4 += 4) {
            float e0 = Eb[(long long)(d4 + 0) * HWP + p];
            float e1 = Eb[(long long)(d4 + 1) * HWP + p];
            float e2 = Eb[(long long)(d4 + 2) * HWP + p];
            float e3 = Eb[(long long)(d4 + 3) * HWP + p];
            esq = fmaf(e0, e0, fmaf(e1, e1, fmaf(e2, e2, fmaf(e3, e3, esq))));
#pragma unroll
            for (int c = 0; c < CC; ++c) {
                v4f m4 = *(const v4f*)&muSh[c * DD + d4];
                cr[c] = fmaf(e0, m4.x, fmaf(e1, m4.y, fmaf(e2, m4.z, fmaf(e3, m4.w, cr[c]))));
            }
        }
#pragma unroll
        for (int c = 0; c < CC; ++c) {
            float mv = (float)Mb[(long long)c * HWP + p];
            float sq = fmaf(-2.0f, cr[c], esq) + muSqSh[c];
            float v = sqrtf(fmaxf(sq, 0.0f) + 1e-12f);
            float r = fmaxf(v * mv - 0.5f, 0.0f);
            vr[c] = fmaf(r, r, vr[c]);
        }
    }
#pragma unroll
    for (int c = 0; c < CC; ++c) atomAddL(&varSh[c], vr[c]);
    __syncthreads();
    if (tx < CC) atomAddG(&varAcc[b * CC + tx], varSh[tx]);
}

__global__ void dl_final_kernel(const float* __restrict__ ws, float* __restrict__ out) {
    if (threadIdx.x == 0 && blockIdx.x == 0) {
        float tot = 0.0f;
        for (int b = 0; b < BB; ++b) {
            float v = 0.0f;
            for (int c = 0; c < CC; ++c) v += ws[WS_VAR + b * CC + c];
            v /= ((float)HWP * (float)CC);
            tot += v + ws[WS_DIST + b] + 0.001f * ws[WS_REG + b];
        }
        out[0] = tot * (1.0f / BB);
    }
}

extern "C" void kernel_launch(void* const* d_in, const int* in_sizes, int n_in,
                              void* d_out, int out_size, void* d_ws, size_t ws_size,
                              hipStream_t stream) {
    const float* E = (const float*)d_in[0];  // [4,32,512,1024] f32
    const int* M = (const int*)d_in[1];      // [4,8,512,1024] i32 (0/1)
    float* out = (float*)d_out;
    float* ws = (float*)d_ws;
    (void)in_sizes; (void)n_in; (void)out_size; (void)ws_size;

    dl_zero_kernel<<<(WS_ZERO_N + 255) / 256, 256, 0, stream>>>(ws);
    dl_pass1_kernel<<<BB * NBLK1, 256, 0, stream>>>(E, M, ws + WS_MUACC, ws + WS_MSUM);
    dl_mu_kernel<<<1, 256, 0, stream>>>(ws);
    dl_pass2_kernel<<<BB * NBLK2, 256, 0, stream>>>(E, M, ws + WS_MU, ws + WS_MUSQ, ws + WS_VAR);
    dl_final_kernel<<<1, 32, 0, stream>>>(ws, out);
}